// QuantumAttentionGate_1176821039145
// MI455X (gfx1250) — compile-verified
//
#include <hip/hip_runtime.h>
#include <hip/hip_bf16.h>
#include <math.h>

typedef __attribute__((ext_vector_type(2))) float v2f;
typedef __attribute__((ext_vector_type(8))) float v8f;

#define NQ      6
#define DIMQ    64
#define DVEC    2048
#define TILE_M  16
#define THREADS 128

// ---------------------------------------------------------------------------
// Kernel A: build BmatT[128][64] = [Re(U)^T ; Im(U)^T] columns, where U is the
// fixed 64x64 unitary from the 2 StronglyEntanglingLayers (Rot + CNOT ring).
// Thread j evolves basis column |j> fully in registers (one-shot, tiny).
// ---------------------------------------------------------------------------
__global__ __launch_bounds__(64) void build_unitary(const float* __restrict__ qw,
                                                    float* __restrict__ BmatT) {
  const int j = threadIdx.x;  // input basis index / column of U
  float re[DIMQ], im[DIMQ];
#pragma unroll
  for (int i = 0; i < DIMQ; ++i) { re[i] = (i == j) ? 1.0f : 0.0f; im[i] = 0.0f; }

#pragma unroll
  for (int l = 0; l < 2; ++l) {
    // per-wire Rot(phi, theta, omega) = RZ(omega) RY(theta) RZ(phi)
#pragma unroll
    for (int w = 0; w < NQ; ++w) {
      const float phi = qw[(l * NQ + w) * 3 + 0];
      const float th  = qw[(l * NQ + w) * 3 + 1];
      const float om  = qw[(l * NQ + w) * 3 + 2];
      const float c   = cosf(0.5f * th), s = sinf(0.5f * th);
      const float ap  = 0.5f * (phi + om), am = 0.5f * (phi - om);
      const float cap = cosf(ap), sap = sinf(ap);
      const float cam = cosf(am), sam = sinf(am);
      // U00 = c e^{-i ap}; U01 = -s e^{+i am}; U10 = s e^{-i am}; U11 = c e^{+i ap}
      const float g00r =  c * cap, g00i = -c * sap;
      const float g01r = -s * cam, g01i = -s * sam;
      const float g10r =  s * cam, g10i = -s * sam;
      const float g11r =  c * cap, g11i =  c * sap;
      const int m = 1 << (5 - w);
#pragma unroll
      for (int i = 0; i < DIMQ; ++i) {
        if ((i & m) == 0) {
          const int i1 = i | m;
          const float ar = re[i], ai = im[i], br = re[i1], bi = im[i1];
          re[i]  = g00r * ar - g00i * ai + g01r * br - g01i * bi;
          im[i]  = g00r * ai + g00i * ar + g01r * bi + g01i * br;
          re[i1] = g10r * ar - g10i * ai + g11r * br - g11i * bi;
          im[i1] = g10r * ai + g10i * ar + g11r * bi + g11i * br;
        }
      }
    }
    // CNOT ring with range r = l + 1
#pragma unroll
    for (int w = 0; w < NQ; ++w) {
      const int t  = (w + l + 1) % NQ;
      const int mc = 1 << (5 - w);
      const int mt = 1 << (5 - t);
#pragma unroll
      for (int i = 0; i < DIMQ; ++i) {
        if ((i & mc) != 0 && (i & mt) == 0) {
          const int i1 = i | mt;
          float tr = re[i]; re[i] = re[i1]; re[i1] = tr;
          float ti = im[i]; im[i] = im[i1]; im[i1] = ti;
        }
      }
    }
  }
  // BmatT[n][k]: n<64 -> Re(U[n][k]), n>=64 -> Im(U[n-64][k]); thread j owns k=j
#pragma unroll
  for (int i = 0; i < DIMQ; ++i) {
    BmatT[i * DIMQ + j]          = re[i];
    BmatT[(DIMQ + i) * DIMQ + j] = im[i];
  }
}

// ---------------------------------------------------------------------------
// Kernel B: fused pipeline, 16 rows per block, 4 waves.
//   GEMM1 (f32 WMMA 16x16x4, K=2048 split over 4 waves) -> angles
//   product-state tile v[16][64] -> GEMM2 (f32 WMMA, K=64, N=128 over 4 waves)
//   -> probs -> PauliZ expvals -> head -> 1 + 0.5*tanh
// ---------------------------------------------------------------------------
__global__ __launch_bounds__(THREADS) void fused_pipeline(
    const float* __restrict__ A,      // [B][2048]
    const float* __restrict__ W1,     // [6][2048]
    const float* __restrict__ b1,     // [6]
    const float* __restrict__ BmatT,  // [128][64]
    const float* __restrict__ W2,     // [16][6]
    const float* __restrict__ b2,     // [16]
    float* __restrict__ out)          // [B][16]
{
  __shared__ float sW1[NQ * DVEC];    // 48 KB; aliased by c2[16][128] after GEMM1
  __shared__ float sred[4 * 16 * 16]; // GEMM1 partials; later epart[16][8][6]
  __shared__ float sca[16 * 8];
  __shared__ float ssa[16 * 8];
  __shared__ float svm[16 * DIMQ];    // product-state tile
  __shared__ float sev[16 * NQ];      // expvals

  const int tid = threadIdx.x;
  const int wv  = tid >> 5;       // wave 0..3
  const int ln  = tid & 31;
  const int m16 = ln & 15;        // M (A row) and N (B col) index per WMMA layout
  const int kh  = ln >> 4;        // K half-select
  const size_t rowBase = (size_t)blockIdx.x * TILE_M;

  // stage W1 into LDS (coalesced, read once per block)
  for (int i = tid; i < NQ * DVEC; i += THREADS) sW1[i] = W1[i];
  __syncthreads();

  // ---- GEMM1: angles_tile[16][6] = A_tile[16][2048] @ W1^T (N padded to 16) ----
  const float bmask = (m16 < NQ) ? 1.0f : 0.0f;
  const int   brow  = (m16 < NQ) ? m16 : (NQ - 1);
  const float* aptr = A + (rowBase + (size_t)m16) * DVEC + wv * 512 + 2 * kh;
  const float* bptr = sW1 + brow * DVEC + wv * 512 + 2 * kh;

  v8f acc = {};
#pragma unroll 8
  for (int s = 0; s < 128; ++s) {
    v2f a = *(const v2f*)(aptr + 4 * s);   // A lane: row m16, K = 4s + 2*kh + {0,1}
    v2f b = *(const v2f*)(bptr + 4 * s);   // B lane: col m16 (clamped), same K pair
    b = b * bmask;                         // zero padded columns 6..15
    acc = __builtin_amdgcn_wmma_f32_16x16x4_f32(
        false, a, false, b, (short)0, acc, false, false);
  }
#pragma unroll
  for (int d = 0; d < 8; ++d)
    sred[wv * 256 + (d + 8 * kh) * 16 + m16] = acc[d];
  __syncthreads();

  // reduce 4 K-partials, add bias, compute half-angle sin/cos
  if (tid < 16 * NQ) {
    const int r = tid / NQ, n = tid % NQ;
    const float ang = sred[r * 16 + n] + sred[256 + r * 16 + n] +
                      sred[512 + r * 16 + n] + sred[768 + r * 16 + n] + b1[n];
    const float h = 0.5f * ang;
    sca[r * 8 + n] = cosf(h);
    ssa[r * 8 + n] = sinf(h);
  }
  __syncthreads();

  // product state: v[r][k] = prod_w (bit_w(k) ? sin : cos), wire 0 = MSB
  for (int idx = tid; idx < 16 * DIMQ; idx += THREADS) {
    const int r = idx >> 6, k = idx & 63;
    float p = 1.0f;
#pragma unroll
    for (int w = 0; w < NQ; ++w)
      p *= ((k >> (5 - w)) & 1) ? ssa[r * 8 + w] : sca[r * 8 + w];
    svm[idx] = p;
  }
  __syncthreads();

  // ---- GEMM2: c2[16][128] = v[16][64] @ BmatT^T ; wave wv owns N in [32wv,32wv+32) ----
  v8f c0 = {}, c1 = {};
  const int n0 = wv * 32 + m16;
  const int n1 = n0 + 16;
  const float* vptr = svm + m16 * DIMQ + 2 * kh;
  const float* bp0  = BmatT + (size_t)n0 * DIMQ + 2 * kh;
  const float* bp1  = BmatT + (size_t)n1 * DIMQ + 2 * kh;
#pragma unroll
  for (int s = 0; s < 16; ++s) {
    v2f a   = *(const v2f*)(vptr + 4 * s);
    v2f bb0 = *(const v2f*)(bp0 + 4 * s);
    v2f bb1 = *(const v2f*)(bp1 + 4 * s);
    c0 = __builtin_amdgcn_wmma_f32_16x16x4_f32(false, a, false, bb0, (short)0, c0, false, false);
    c1 = __builtin_amdgcn_wmma_f32_16x16x4_f32(false, a, false, bb1, (short)0, c1, false, false);
  }
  float* c2 = sW1;  // alias: W1 staging dead after GEMM1 (barrier-separated)
#pragma unroll
  for (int d = 0; d < 8; ++d) {
    const int m = d + 8 * kh;
    c2[m * 128 + n0] = c0[d];
    c2[m * 128 + n1] = c1[d];
  }
  __syncthreads();

  // probs = wr^2 + wi^2 ; signed partial sums over 8-wide k slices
  {
    const int r = tid >> 3, kg = tid & 7;
    float e[NQ] = {};
#pragma unroll
    for (int q = 0; q < 8; ++q) {
      const int k = kg * 8 + q;
      const float wr = c2[r * 128 + k];
      const float wi = c2[r * 128 + 64 + k];
      const float p = wr * wr + wi * wi;
#pragma unroll
      for (int w = 0; w < NQ; ++w)
        e[w] += ((k >> (5 - w)) & 1) ? -p : p;
    }
#pragma unroll
    for (int w = 0; w < NQ; ++w)
      sred[(r * 8 + kg) * NQ + w] = e[w];  // sred reused as epart
  }
  __syncthreads();

  if (tid < 16 * NQ) {
    const int r = tid / NQ, w = tid % NQ;
    float e = 0.0f;
#pragma unroll
    for (int kg = 0; kg < 8; ++kg) e += sred[(r * 8 + kg) * NQ + w];
    sev[r * NQ + w] = e;
  }
  __syncthreads();

  // head: out = 1 + 0.5 * tanh(expvals @ W2^T + b2); 2 outputs per thread
  {
    const int r = tid >> 3;
    const int hbase = (tid & 7) * 2;
#pragma unroll
    for (int hh = 0; hh < 2; ++hh) {
      const int h = hbase + hh;
      float a2 = b2[h];
#pragma unroll
      for (int w = 0; w < NQ; ++w) a2 += sev[r * NQ + w] * W2[h * NQ + w];
      out[(rowBase + r) * 16 + h] = 1.0f + 0.5f * tanhf(a2);
    }
  }
}

extern "C" void kernel_launch(void* const* d_in, const int* in_sizes, int n_in,
                              void* d_out, int out_size, void* d_ws, size_t ws_size,
                              hipStream_t stream) {
  const float* sv = (const float*)d_in[0];
  const float* W1 = (const float*)d_in[1];
  const float* b1 = (const float*)d_in[2];
  const float* qw = (const float*)d_in[3];
  const float* W2 = (const float*)d_in[4];
  const float* b2 = (const float*)d_in[5];
  float* out      = (float*)d_out;
  float* BmatT    = (float*)d_ws;  // 128*64 floats = 32 KB

  const int B = in_sizes[0] / DVEC;  // 16384
  build_unitary<<<1, 64, 0, stream>>>(qw, BmatT);
  fused_pipeline<<<B / TILE_M, THREADS, 0, stream>>>(sv, W1, b1, BmatT, W2, b2, out);
}